// Decoding_model_10230612099665
// MI455X (gfx1250) — compile-verified
//
#include <hip/hip_runtime.h>
#include <hip/hip_bf16.h>
#include <stdint.h>

#define N_COLS   2048
#define N_ROWS   1024
#define N_B      8
#define ROW_W    6
#define N_ITERS  10
#define BIGV     1.0e30f
#define TOT      (N_B * N_COLS)   /* 16384 */

typedef unsigned int u32x4 __attribute__((ext_vector_type(4)));
typedef int          i32x8 __attribute__((ext_vector_type(8)));
typedef int          i32x4 __attribute__((ext_vector_type(4)));

// ---------------------------------------------------------------------------
// TDM: DMA `nelem` f32 from global into LDS at (lds_base + off_elems*4).
// D# per CDNA5 ISA ch.8: group0 = {count=1, lds_addr, ga_lo, ga_hi|type=2};
// group1: data_size=4B, tensor nelem x 1, tile nelem x 1, dim0 stride nelem.
// Groups 2/3 zero (<=2D). EXEC ignored: one DMA per issuing wave.
// ---------------------------------------------------------------------------
__device__ __forceinline__ void tdm_load_f32(void* lds_base, const float* gsrc,
                                             unsigned off_elems, unsigned nelem) {
  uint32_t lds_off = (uint32_t)(uintptr_t)lds_base + off_elems * 4u;
  uint64_t ga      = (uint64_t)(uintptr_t)(gsrc + off_elems);
  u32x4 g0 = { 1u,                                       // count = 1, user mode
               lds_off,                                  // lds_addr (bytes)
               (uint32_t)(ga & 0xFFFFFFFFull),           // global_addr[31:0]
               ((uint32_t)(ga >> 32) & 0x01FFFFFFu) | (2u << 30) }; // hi | type=2
  i32x8 g1 = { (int)(2u << 16),          // data_size = 2 (4 bytes), no multicast
               (int)(nelem << 16),       // tensor_dim0[15:0] @ bits 63:48
               (int)((nelem >> 16) | (1u << 16)),   // dim0[31:16] + tensor_dim1=1
               (int)(nelem << 16),       // tile_dim0 @ bits 127:112 (nelem<=65535)
               1,                        // tile_dim1 = 1
               (int)nelem,               // tensor_dim0_stride[31:0]
               0, 0 };
  i32x4 z4 = { 0, 0, 0, 0 };
#if __has_include(<hip/amd_detail/amd_gfx1250_TDM.h>)
  i32x8 z8 = { 0, 0, 0, 0, 0, 0, 0, 0 };
  __builtin_amdgcn_tensor_load_to_lds(g0, g1, z4, z4, z8, 0);
#else
  __builtin_amdgcn_tensor_load_to_lds(g0, g1, z4, z4, 0);
#endif
  __builtin_amdgcn_s_wait_tensorcnt(0);
}

// ---------------------------------------------------------------------------
// Persistent single-WGP normalized min-sum LDPC decoder (latency-bound problem:
// ~5 MFLOP / <1 MB hot state total -> keep everything LDS/register resident).
// LDS: total[8*2048] f32 (64KB) + compact check->var messages per (batch,row):
// {sign/use-min2 bits ; gamma*min1 ; gamma*min2} (96KB).
// Thread tid owns check row r=tid for all 8 batches, and output elements
// [tid*16, tid*16+16).
// ---------------------------------------------------------------------------
__global__ __launch_bounds__(1024) void ldpc_decode(const float* __restrict__ soft,
                                                    const int*   __restrict__ labels,
                                                    const float* __restrict__ normalizer,
                                                    const int*   __restrict__ rowidx,
                                                    float*       __restrict__ out) {
  __shared__ float    s_total[TOT];            // 64 KB
  __shared__ uint32_t s_bits[N_B * N_ROWS];    // 32 KB
  __shared__ float    s_m1[N_B * N_ROWS];      // 32 KB
  __shared__ float    s_m2[N_B * N_ROWS];      // 32 KB

  const int tid = threadIdx.x;

  // total := soft_input via TDM, issued immediately (wave 0; EXEC ignored).
  if (tid == 0) tdm_load_f32(s_total, soft, 0u, (unsigned)TOT);

  const float gamma = log1pf(expf(normalizer[0]));   // softplus(normalizer)

  int cidx[ROW_W];
#pragma unroll
  for (int e = 0; e < ROW_W; ++e) cidx[e] = rowidx[tid * 8 + e];

  // cv := 0
#pragma unroll
  for (int k = 0; k < 8; ++k) {
    int i = tid + k * 1024;
    s_bits[i] = 0u; s_m1[i] = 0.0f; s_m2[i] = 0.0f;
  }

  // labels -> regs + labels output (cast to float)
  float lab[16];
#pragma unroll
  for (int j = 0; j < 16; ++j) {
    int i = tid * 16 + j;
    float lf = (float)labels[i];
    lab[j] = lf;
    out[11 * TOT + i] = lf;
  }

  __syncthreads();   // TDM data visible (wave0 did s_wait_tensorcnt pre-barrier)

  // stacked[0] = soft_input
#pragma unroll
  for (int j = 0; j < 16; ++j) {
    int i = tid * 16 + j;
    out[i] = s_total[i];
  }

  float loss = 0.0f;

#pragma unroll 1
  for (int it = 0; it < N_ITERS; ++it) {
    // ---- Phase A: per-row min-sum update (vc = total - cv_old) ----
#pragma unroll 1
    for (int b = 0; b < N_B; ++b) {
      const int cvi = b * N_ROWS + tid;
      const uint32_t ob = s_bits[cvi];
      const float om1 = s_m1[cvi], om2 = s_m2[cvi];
      float vc[ROW_W], av[ROW_W];
#pragma unroll
      for (int e = 0; e < ROW_W; ++e) {
        float c = ((ob >> (6 + e)) & 1u) ? om2 : om1;   // this edge was a min?
        c = ((ob >> e) & 1u) ? -c : c;                  // extrinsic sign
        vc[e] = s_total[b * N_COLS + cidx[e]] - c;
      }
      float mn = 3.4e38f, mn2 = 3.4e38f;
      uint32_t nneg = 0u; bool anyzero = false;
#pragma unroll
      for (int e = 0; e < ROW_W; ++e) {
        float a = fminf(fabsf(vc[e]), BIGV);            // clip(|vc|, 0, BIG)
        av[e] = a;
        anyzero = anyzero || (vc[e] == 0.0f);           // jnp.sign(0)=0 kills row
        nneg += (vc[e] < 0.0f) ? 1u : 0u;
        float nm2 = (a < mn2) ? a : mn2;
        if (a < mn) { mn2 = mn; mn = a; } else { mn2 = nm2; }
      }
      const uint32_t par = nneg & 1u;
      uint32_t nb = 0u;
#pragma unroll
      for (int e = 0; e < ROW_W; ++e) {
        uint32_t sb = par ^ ((vc[e] < 0.0f) ? 1u : 0u); // row_sign * own sign
        uint32_t ub = (av[e] <= mn) ? 1u : 0u;          // ref: !(vc_abs>mn) -> mn2
        nb |= (sb << e) | (ub << (6 + e));
      }
      s_bits[cvi] = nb;
      s_m1[cvi] = anyzero ? 0.0f : gamma * mn;
      s_m2[cvi] = anyzero ? 0.0f : gamma * mn2;
    }
    __syncthreads();

    // ---- Phase B: total := soft_input. Two statically distinct TDM sites,
    //      issued concurrently by wave 0 (first half) and wave 1 (second half).
    if (tid == 0) {
      tdm_load_f32(s_total, soft, 0u, (unsigned)(TOT / 2));
    } else if (tid == 32) {
      tdm_load_f32(s_total, soft, (unsigned)(TOT / 2), (unsigned)(TOT / 2));
    }
    __syncthreads();

    // ---- Phase C: scatter cv_new into total (ds_add_f32 atomics) ----
#pragma unroll 1
    for (int b = 0; b < N_B; ++b) {
      const int cvi = b * N_ROWS + tid;
      const uint32_t nb = s_bits[cvi];
      const float m1 = s_m1[cvi], m2 = s_m2[cvi];
#pragma unroll
      for (int e = 0; e < ROW_W; ++e) {
        float v = ((nb >> (6 + e)) & 1u) ? m2 : m1;
        v = ((nb >> e) & 1u) ? -v : v;
        atomicAdd(&s_total[b * N_COLS + cidx[e]], v);
      }
    }
    __syncthreads();

    // ---- Phase D: emit soft_out slab + sigmoid cross-entropy loss ----
#pragma unroll
    for (int j = 0; j < 16; ++j) {
      int i = tid * 16 + j;
      float t = s_total[i];
      out[(it + 1) * TOT + i] = t;
      float z = -t;
      loss += fmaxf(z, 0.0f) - z * lab[j] + log1pf(expf(-fabsf(z)));
    }
    __syncthreads();
  }

  // ---- loss reduction (reuse s_total as scratch) ----
  s_total[tid] = loss;
  __syncthreads();
#pragma unroll 1
  for (int s = 512; s > 0; s >>= 1) {
    if (tid < s) s_total[tid] += s_total[tid + s];
    __syncthreads();
  }
  if (tid == 0) out[12 * TOT] = s_total[0];
}

// ---------------------------------------------------------------------------
// Extract the 6 column indices of each parity-check row from dense H. Edge
// order within a row is irrelevant to the decoder math (min/min2, sign parity
// and the scatter-sum are order independent), so an atomic counter is fine.
// ---------------------------------------------------------------------------
__global__ __launch_bounds__(256) void ldpc_extract_rows(const int* __restrict__ H,
                                                         int* __restrict__ rowidx) {
  __shared__ int cnt;
  const int r = blockIdx.x;
  if (threadIdx.x == 0) cnt = 0;
  __syncthreads();
  for (int j = threadIdx.x; j < N_COLS; j += 256) {
    if (H[r * N_COLS + j] != 0) {
      int p = atomicAdd(&cnt, 1);
      if (p < 8) rowidx[r * 8 + p] = j;
    }
  }
}

// ---------------------------------------------------------------------------
// Inputs (setup_inputs order): soft_input f32[8*2048], labels i32[8*2048],
// H i32[1024*2048], normalizer f32[1].
// Output: [stacked f32[11*16384] | labels-as-f32[16384] | loss f32[1]].
// ---------------------------------------------------------------------------
extern "C" void kernel_launch(void* const* d_in, const int* in_sizes, int n_in,
                              void* d_out, int out_size, void* d_ws, size_t ws_size,
                              hipStream_t stream) {
  const float* soft       = (const float*)d_in[0];
  const int*   labels     = (const int*)d_in[1];
  const int*   H          = (const int*)d_in[2];
  const float* normalizer = (const float*)d_in[3];
  int* rowidx = (int*)d_ws;                        // 1024 rows * 8 ints = 32 KB

  ldpc_extract_rows<<<N_ROWS, 256, 0, stream>>>(H, rowidx);
  ldpc_decode<<<1, 1024, 0, stream>>>(soft, labels, normalizer, rowidx, (float*)d_out);
}